// GNNAutoEncoder_22789096472977
// MI455X (gfx1250) — compile-verified
//
#include <hip/hip_runtime.h>
#include <hip/hip_bf16.h>

typedef __attribute__((ext_vector_type(2))) float v2f;
typedef __attribute__((ext_vector_type(8))) float v8f;

#define D_FEAT 256
#define HIDDEN 256
#define OUTC   128

// ---------------------------------------------------------------------------
// D = A(16x4) * B(4x16) + C  in fp32 via V_WMMA_F32_16X16X4_F32
// A: lane 0-15 -> M=lane, V0=K0,V1=K1 ; lane 16-31 -> M=lane-16, V0=K2,V1=K3
// B: lane 0-15 -> N=lane, V0=K0,V1=K1 ; lane 16-31 -> N=lane-16, V0=K2,V1=K3
// C/D: VGPR r -> M = r + 8*(lane>=16), N = lane&15
// ---------------------------------------------------------------------------
__device__ __forceinline__ v8f wmma_f32(v2f a, v2f b, v8f c) {
    return __builtin_amdgcn_wmma_f32_16x16x4_f32(false, a, false, b, (short)0, c,
                                                 false, false);
}

// ---------------------------------------------------------------------------
__global__ void zero_f32(float* __restrict__ p, long long n) {
    long long i = (long long)blockIdx.x * blockDim.x + threadIdx.x;
    long long stride = (long long)gridDim.x * blockDim.x;
    for (; i < n; i += stride) p[i] = 0.0f;
}

// ---------------------------------------------------------------------------
// One 256-wide feature row per edge: agg[dst] += feat[src]; cnt[dst] += 1
// ---------------------------------------------------------------------------
__global__ void scatter_add(const float* __restrict__ feat,
                            const int* __restrict__ src,
                            const int* __restrict__ dst,
                            float* __restrict__ agg,
                            float* __restrict__ cnt,
                            long long E, int edgesPerBlock) {
    int col = threadIdx.x;  // 0..255
    long long base = (long long)blockIdx.x * edgesPerBlock;
    for (int i = 0; i < edgesPerBlock; ++i) {
        long long e = base + i;
        if (e >= E) return;
        long long s = (long long)src[e];
        long long d = (long long)dst[e];
        atomicAdd(agg + d * D_FEAT + col, feat[s * D_FEAT + col]);
        if (cnt != nullptr && col == 0) atomicAdd(cnt + d, 1.0f);
    }
}

// ---------------------------------------------------------------------------
// out[m, o] = act( (agg[m,:]/max(cnt[m],1)) @ Wl[o,:] + bl[o] + self[m,:] @ Wr[o,:] )
// One wave computes a 16(node) x 16(outchan) tile, K = 256.
// ---------------------------------------------------------------------------
__global__ void sage_wmma(const float* __restrict__ agg,
                          const float* __restrict__ cnt,
                          const float* __restrict__ selfF,
                          const float* __restrict__ Wl,
                          const float* __restrict__ bl,
                          const float* __restrict__ Wr,
                          float* __restrict__ outF,
                          int nTiles, int oTiles, int O, int doRelu) {
    int lane = threadIdx.x & 31;
    int wid  = threadIdx.x >> 5;
    long long gw = (long long)blockIdx.x * (blockDim.x >> 5) + wid;
    long long nt = gw / oTiles;
    int ot = (int)(gw % oTiles);
    if (nt >= nTiles) return;  // wave-uniform

    int rlane = lane & 15;
    int half  = lane >> 4;
    int koff  = half * 2;

    long long m = nt * 16 + rlane;       // node row this lane loads
    int       o = ot * 16 + rlane;       // out channel this lane loads / stores

    float deg = cnt[m];
    float inv = 1.0f / fmaxf(deg, 1.0f);

    const float* aRow  = agg   + m * (long long)D_FEAT + koff;
    const float* sRow  = selfF + m * (long long)D_FEAT + koff;
    const float* wlRow = Wl    + (long long)o * D_FEAT + koff;
    const float* wrRow = Wr    + (long long)o * D_FEAT + koff;

    v8f acc = {};
    // mean-aggregated neighbor term
    #pragma unroll 4
    for (int k = 0; k < D_FEAT; k += 4) {
        v2f a = *(const v2f*)(aRow + k);
        a *= inv;
        v2f b = *(const v2f*)(wlRow + k);
        acc = wmma_f32(a, b, acc);
    }
    // self (root) term
    #pragma unroll 4
    for (int k = 0; k < D_FEAT; k += 4) {
        v2f a = *(const v2f*)(sRow + k);
        v2f b = *(const v2f*)(wrRow + k);
        acc = wmma_f32(a, b, acc);
    }

    float bias = bl[o];
    #pragma unroll
    for (int r = 0; r < 8; ++r) {
        float v = acc[r] + bias;
        if (doRelu) v = fmaxf(v, 0.0f);
        long long row = nt * 16 + r + 8 * half;
        outF[row * O + o] = v;
    }
}

// ---------------------------------------------------------------------------
// Edge decoder: zc = [z[src], z[dst]] (256), hd = relu(zc @ Wd1.T + bd1) (128),
// out = hd . Wd2 + bd2.  One wave -> 16 edges x all 128 hidden channels
// (8 accumulators).  Wd1 staged in LDS in 64-wide K chunks.
// ---------------------------------------------------------------------------
#define DEC_PITCH 68  // 68 dwords/row -> conflict-free 64-bit LDS reads

__global__ void edge_dec_wmma(const float* __restrict__ z,
                              const int* __restrict__ src,
                              const int* __restrict__ dst,
                              const float* __restrict__ Wd1,
                              const float* __restrict__ bd1,
                              const float* __restrict__ Wd2,
                              const float* __restrict__ bd2,
                              float* __restrict__ out,
                              long long eTiles) {
    __shared__ float sW[OUTC * DEC_PITCH];

    int tid   = threadIdx.x;
    int lane  = tid & 31;
    int wid   = tid >> 5;
    int rlane = lane & 15;
    int half  = lane >> 4;
    int koff  = half * 2;

    long long et = (long long)blockIdx.x * 8 + wid;
    if (et >= eTiles) return;  // wave-uniform (grid is exact anyway)

    long long e = et * 16 + rlane;
    long long s = (long long)src[e];
    long long d = (long long)dst[e];
    const float* zs = z + s * (long long)OUTC;
    const float* zd = z + d * (long long)OUTC;

    v8f acc[8];
    #pragma unroll
    for (int ot = 0; ot < 8; ++ot) acc[ot] = (v8f){};

    for (int kc = 0; kc < 2 * OUTC; kc += 64) {
        // cooperatively stage Wd1[:, kc:kc+64] into LDS (128 rows x 64 cols)
        for (int idx = tid; idx < OUTC * 32; idx += 256) {
            int o  = idx >> 5;
            int kk = (idx & 31) * 2;
            v2f w = *(const v2f*)(Wd1 + (long long)o * (2 * OUTC) + kc + kk);
            sW[o * DEC_PITCH + kk]     = w.x;
            sW[o * DEC_PITCH + kk + 1] = w.y;
        }
        __syncthreads();

        const float* zp = (kc < OUTC) ? zs : zd;  // concat([z_src, z_dst])
        int cbase = (kc & (OUTC - 1)) + koff;

        for (int k = 0; k < 64; k += 4) {
            v2f a = *(const v2f*)(zp + cbase + k);
            int kl = k + koff;
            #pragma unroll
            for (int ot = 0; ot < 8; ++ot) {
                v2f b = *(const v2f*)(&sW[(ot * 16 + rlane) * DEC_PITCH + kl]);
                acc[ot] = wmma_f32(a, b, acc[ot]);
            }
        }
        __syncthreads();
    }

    // Epilogue: relu(+bd1), dot with Wd2, reduce across the 16 N-lanes.
    float part[8];
    #pragma unroll
    for (int r = 0; r < 8; ++r) part[r] = 0.0f;
    #pragma unroll
    for (int ot = 0; ot < 8; ++ot) {
        int o = ot * 16 + rlane;
        float bb = bd1[o];
        float ww = Wd2[o];
        #pragma unroll
        for (int r = 0; r < 8; ++r)
            part[r] += fmaxf(acc[ot][r] + bb, 0.0f) * ww;
    }
    #pragma unroll
    for (int mask = 1; mask <= 8; mask <<= 1) {
        #pragma unroll
        for (int r = 0; r < 8; ++r)
            part[r] += __shfl_xor(part[r], mask, 32);
    }
    if (rlane == 0) {
        float b2 = bd2[0];
        long long ebase = et * 16 + (long long)half * 8;  // half0: rows 0-7, half1: 8-15
        #pragma unroll
        for (int r = 0; r < 8; ++r) out[ebase + r] = part[r] + b2;
    }
}

// ---------------------------------------------------------------------------
extern "C" void kernel_launch(void* const* d_in, const int* in_sizes, int n_in,
                              void* d_out, int out_size, void* d_ws, size_t ws_size,
                              hipStream_t stream) {
    const float* x    = (const float*)d_in[0];
    const int*   eidx = (const int*)d_in[1];   // (2, E), int32 per harness contract
    const float* W1l  = (const float*)d_in[2];
    const float* b1l  = (const float*)d_in[3];
    const float* W1r  = (const float*)d_in[4];
    const float* W2l  = (const float*)d_in[5];
    const float* b2l  = (const float*)d_in[6];
    const float* W2r  = (const float*)d_in[7];
    const float* Wd1  = (const float*)d_in[8];
    const float* bd1  = (const float*)d_in[9];
    const float* Wd2  = (const float*)d_in[10];
    const float* bd2  = (const float*)d_in[11];
    float* out = (float*)d_out;

    const long long N = in_sizes[0] / D_FEAT;   // 100000
    const long long E = in_sizes[1] / 2;        // 1600000
    const int* srcI = eidx;
    const int* dstI = eidx + E;

    // workspace layout (fp32): cnt[N] | agg[N*256] | h[N*256] | z[N*128]
    float* cnt = (float*)d_ws;
    float* agg = cnt + N;
    float* h   = agg + N * D_FEAT;
    float* zz  = h + N * D_FEAT;

    const int ZB = 4096;
    // ---- layer 1: zero cnt+agg (contiguous), scatter x, SAGE conv -> h ----
    zero_f32<<<ZB, 256, 0, stream>>>(cnt, N + N * (long long)D_FEAT);
    {
        int epb = 16;
        long long blocks = (E + epb - 1) / epb;
        scatter_add<<<(int)blocks, 256, 0, stream>>>(x, srcI, dstI, agg, cnt, E, epb);
    }
    {
        int nTiles = (int)(N / 16);                  // 6250
        int oTiles = HIDDEN / 16;                    // 16
        long long waves = (long long)nTiles * oTiles;
        int blocks = (int)((waves + 7) / 8);
        sage_wmma<<<blocks, 256, 0, stream>>>(agg, cnt, x, W1l, b1l, W1r, h,
                                              nTiles, oTiles, HIDDEN, 1);
    }
    // ---- layer 2: re-zero agg, scatter h, SAGE conv -> z ----
    zero_f32<<<ZB, 256, 0, stream>>>(agg, N * (long long)D_FEAT);
    {
        int epb = 16;
        long long blocks = (E + epb - 1) / epb;
        scatter_add<<<(int)blocks, 256, 0, stream>>>(h, srcI, dstI, agg, nullptr, E, epb);
    }
    {
        int nTiles = (int)(N / 16);
        int oTiles = OUTC / 16;                      // 8
        long long waves = (long long)nTiles * oTiles;
        int blocks = (int)((waves + 7) / 8);
        sage_wmma<<<blocks, 256, 0, stream>>>(agg, cnt, h, W2l, b2l, W2r, zz,
                                              nTiles, oTiles, OUTC, 0);
    }
    // ---- edge decoder ----
    {
        long long eTiles = E / 16;                   // 100000
        int blocks = (int)((eTiles + 7) / 8);        // 12500
        edge_dec_wmma<<<blocks, 256, 0, stream>>>(zz, srcI, dstI, Wd1, bd1, Wd2, bd2,
                                                  out, eTiles);
    }
}